// Jxy_val_34557306864072
// MI455X (gfx1250) — compile-verified
//
#include <hip/hip_runtime.h>
#include <math.h>

typedef __attribute__((ext_vector_type(2))) float v2f;
typedef __attribute__((ext_vector_type(8))) float v8f;

#define NALL  1024
#define NH    512
#define NCOL  8
#define BATCH 128
#define PITCH 520          // 512 cols + pad; 520 % 64 == 8 -> conflict-free LDS banks
#define WROWS 17           // rows 0..7 = un^T, 8..15 = vn^T, 16 = zeros
#define WELEMS (WROWS * PITCH)

// ---------------------------------------------------------------------------
// Prep: joint column norms of [u;v], write normalized transposed weight rows
// (plus a zero row) to workspace, and zero the output accumulator.
// ---------------------------------------------------------------------------
__global__ __launch_bounds__(256) void prep_kernel(const float* __restrict__ u,
                                                   const float* __restrict__ v,
                                                   float* __restrict__ w,
                                                   float* __restrict__ out) {
  __shared__ float nrm[NCOL];
  __shared__ float rn[NCOL];
  const int t = threadIdx.x;
  if (t < NCOL) nrm[t] = 0.f;
  for (int i = t; i < WELEMS; i += 256) w[i] = 0.f;
  for (int i = t; i < BATCH * 4 * NCOL; i += 256) out[i] = 0.f;
  __syncthreads();

  float s[NCOL];
#pragma unroll
  for (int c = 0; c < NCOL; ++c) s[c] = 0.f;
  for (int r = t; r < NH; r += 256) {
#pragma unroll
    for (int c = 0; c < NCOL; ++c) {
      float a = u[r * NCOL + c]; s[c] += a * a;
      float b = v[r * NCOL + c]; s[c] += b * b;
    }
  }
#pragma unroll
  for (int c = 0; c < NCOL; ++c) atomicAdd(&nrm[c], s[c]);
  __syncthreads();
  if (t < NCOL) rn[t] = 1.0f / sqrtf(nrm[t]);
  __syncthreads();

  // w layout: row c  (c<8)  : un^T[c][0..511]
  //           row 8+c       : vn^T[c][0..511]
  //           row 16        : zeros
  for (int idx = t; idx < NH * NCOL; idx += 256) {
    const int r = idx >> 3, c = idx & 7;
    w[c * PITCH + r]          = u[idx] * rn[c];
    w[(NCOL + c) * PITCH + r] = v[idx] * rn[c];
  }
}

// ---------------------------------------------------------------------------
// Main: one wave per 16-row tile of J.  acc (16x16 f32) = J_tile @ W16 via
// 256 x v_wmma_f32_16x16x4_f32, then weighted row-reduction to 4 quadrants.
// ---------------------------------------------------------------------------
__global__ __launch_bounds__(256) void jquad_kernel(const float* __restrict__ x,
                                                    const float* __restrict__ w,
                                                    float* __restrict__ out) {
  __shared__ float lw[WELEMS];
  __shared__ float out32[32];
  const int t = threadIdx.x;
  for (int i = t; i < WELEMS; i += 256) lw[i] = w[i];
  if (t < 32) out32[t] = 0.f;
  __syncthreads();

  const int b     = blockIdx.x >> 3;   // 0..127
  const int chunk = blockIdx.x & 7;    // 8 row-chunks of 128 rows
  const int wave  = t >> 5;            // 8 waves per block
  const int lane  = t & 31;
  const int tile  = chunk * 8 + wave;  // 0..63  (16 rows each)
  const int r0    = tile * 16;

  const int  m     = lane & 15;        // A: row M / B,C,D: col N
  const int  khalf = lane >> 4;        // 0 -> K{0,1}, 1 -> K{2,3}
  const int  c     = m & 7;
  const bool laneU = (m < 8);          // cols 0..7 = Y_u, 8..15 = Y_v

  // A operand: lane holds J[r0+m, j + 2*khalf + {0,1}]  (contiguous float2)
  const float* rowp =
      x + (size_t)b * (NALL * NALL) + (size_t)(r0 + m) * NALL + khalf * 2;

  // B operand sources (branchless: zero-row for the dead half)
  const v2f* bu = (const v2f*)&lw[(laneU ? c : 16) * PITCH];        // j <  512
  const v2f* bv = (const v2f*)&lw[(laneU ? 16 : (8 + c)) * PITCH];  // j >= 512

  v8f acc = {};

  // j in [0, 512): accumulate Y_u into cols 0..7 (cols 8..15 see zeros)
#pragma unroll 8
  for (int j = 0; j < NH; j += 4) {
    v2f a  = __builtin_nontemporal_load((const v2f*)(rowp + j));
    v2f bb = bu[(j >> 1) + khalf];
    acc = __builtin_amdgcn_wmma_f32_16x16x4_f32(false, a, false, bb,
                                                (short)0, acc, false, false);
  }
  // j in [512, 1024): accumulate Y_v into cols 8..15
#pragma unroll 8
  for (int j = 0; j < NH; j += 4) {
    v2f a  = __builtin_nontemporal_load((const v2f*)(rowp + NH + j));
    v2f bb = bv[(j >> 1) + khalf];
    acc = __builtin_amdgcn_wmma_f32_16x16x4_f32(false, a, false, bb,
                                                (short)0, acc, false, false);
  }

  // Epilogue: D layout -> lane holds col n=m, rows i = r0 + q + 8*khalf.
  // out = [uJu | vJu | uJv | vJv], 8 cols each.
  const int i0 = r0 + khalf * 8;
  const float* srow = (i0 < NH) ? &lw[c * PITCH + i0]
                                : &lw[(NCOL + c) * PITCH + (i0 - NH)];
  float partial = 0.f;
#pragma unroll
  for (int q = 0; q < 8; ++q) partial += srow[q] * acc[q];

  const int qbase = ((i0 < NH) ? 0 : 8) + (laneU ? 0 : 16);
  atomicAdd(&out32[qbase + c], partial);
  __syncthreads();
  if (t < 32) atomicAdd(&out[b * 32 + t], out32[t]);
}

// ---------------------------------------------------------------------------
extern "C" void kernel_launch(void* const* d_in, const int* in_sizes, int n_in,
                              void* d_out, int out_size, void* d_ws, size_t ws_size,
                              hipStream_t stream) {
  (void)in_sizes; (void)n_in; (void)out_size; (void)ws_size;
  const float* x = (const float*)d_in[0];
  const float* u = (const float*)d_in[1];
  const float* v = (const float*)d_in[2];
  float* out = (float*)d_out;
  float* w   = (float*)d_ws;   // needs 17*520*4 = 35,360 bytes

  prep_kernel<<<1, 256, 0, stream>>>(u, v, w, out);
  jquad_kernel<<<BATCH * 8, 256, 0, stream>>>(x, w, out);
}